// BD_59502476919453
// MI455X (gfx1250) — compile-verified
//
#include <hip/hip_runtime.h>
#include <math.h>

typedef float v2f __attribute__((ext_vector_type(2)));
typedef float v4f __attribute__((ext_vector_type(4)));
typedef float v8f __attribute__((ext_vector_type(8)));

#define GRP   16
#define CHW   200704            // 64*56*56 columns per batch image
#define NB    16
#define MCOLS (NB * CHW)        // 3,211,264 total columns
#define EPSW  1e-5f

// Decomposition shared by both streaming kernels:
//   1792 blocks x 256 threads = 14336 waves; 896 waves per n (14336/16);
//   each wave: 14 tiles x 16 columns = 224 contiguous columns; 896*224 == CHW.
#define BLOCKS      1792
#define WAVES_PER_N 896
#define TILES_PER_W 14

// ws layout (floats): [0..255] gram, [256..271] rowsum, [272..527] wm = L^{-1}

__global__ void BD_ws_zero(float* __restrict__ ws) {
    int i = threadIdx.x;
    if (i < 272) ws[i] = 0.0f;
}

// ---------------------------------------------------------------------------
// Pass 1: Gram = X X^T and row sums.
// K-slot assignment (free choice, K is a symmetric summation index):
//   lanes 0-15  (hold K=0,1 of each WMMA): columns [j .. j+8)
//   lanes 16-31 (hold K=2,3 of each WMMA): columns [j+8 .. j+16)
// -> every lane loads 2x b128 (32 B contiguous), 4 WMMAs cover 16 columns.
// B == A registers for the symmetric product X_chunk * X_chunk^T.
// ---------------------------------------------------------------------------
__global__ void BD_gram(const float* __restrict__ x, float* __restrict__ ws) {
    __shared__ float sgram[256];
    __shared__ float ssum[16];
    const int tid = threadIdx.x;
    for (int i = tid; i < 256; i += blockDim.x) sgram[i] = 0.0f;
    if (tid < 16) ssum[tid] = 0.0f;
    __syncthreads();

    const int lane = tid & 31;
    const int wib  = __builtin_amdgcn_readfirstlane(tid >> 5);   // SGPR wave id
    const int wave = blockIdx.x * 8 + wib;                       // scalar
    const int n    = wave / WAVES_PER_N;                         // scalar
    const int tb   = wave - n * WAVES_PER_N;                     // scalar

    const int g  = lane & 15;
    const int hk = (lane < 16) ? 0 : 8;      // this half-wave's column sub-block

    const float* p = x + (size_t)(n * GRP + g) * CHW
                       + (size_t)tb * (TILES_PER_W * 16) + hk;

    v8f acc0 = {}, acc1 = {};
    float s = 0.0f;

#pragma unroll 2
    for (int it = 0; it < TILES_PER_W; ++it) {
        __builtin_prefetch(p + 64, 0, 0);                 // global_prefetch, 256 B ahead
        v4f q0 = *(const v4f*)(p);                        // b128: 4 columns
        v4f q1 = *(const v4f*)(p + 4);                    // b128: next 4 columns
        v2f a0 = {q0.x, q0.y};
        v2f a1 = {q0.z, q0.w};
        v2f a2 = {q1.x, q1.y};
        v2f a3 = {q1.z, q1.w};
        acc0 = __builtin_amdgcn_wmma_f32_16x16x4_f32(false, a0, false, a0, (short)0, acc0, false, false);
        acc1 = __builtin_amdgcn_wmma_f32_16x16x4_f32(false, a1, false, a1, (short)0, acc1, false, false);
        acc0 = __builtin_amdgcn_wmma_f32_16x16x4_f32(false, a2, false, a2, (short)0, acc0, false, false);
        acc1 = __builtin_amdgcn_wmma_f32_16x16x4_f32(false, a3, false, a3, (short)0, acc1, false, false);
        s += ((q0.x + q0.y) + (q0.z + q0.w)) + ((q1.x + q1.y) + (q1.z + q1.w));
        p += 16;
    }
    acc0 = acc0 + acc1;

    // acc0[r]: lanes 0-15 -> Gram[r][lane], lanes 16-31 -> Gram[r+8][lane-16]
#pragma unroll
    for (int r = 0; r < 8; ++r) {
        int M = r + ((lane < 16) ? 0 : 8);
        atomicAdd(&sgram[M * 16 + (lane & 15)], acc0[r]);
    }
    atomicAdd(&ssum[g], s);
    __syncthreads();
    for (int i = tid; i < 256; i += blockDim.x) atomicAdd(&ws[i], sgram[i]);
    if (tid < 16) atomicAdd(&ws[256 + tid], ssum[tid]);
}

// ---------------------------------------------------------------------------
// Pass 2: tiny serial 16x16 linear algebra -> wm = L^{-1}
// ---------------------------------------------------------------------------
__global__ void BD_chol(float* __restrict__ ws) {
    if (threadIdx.x != 0 || blockIdx.x != 0) return;
    const float invm = 1.0f / (float)MCOLS;
    float mean[16], S[16][16], L[16][16];
    for (int i = 0; i < 16; ++i) mean[i] = ws[256 + i] * invm;
    for (int i = 0; i < 16; ++i)
        for (int j = 0; j < 16; ++j) {
            float v = ws[i * 16 + j] * invm - mean[i] * mean[j];
            if (i == j) v += EPSW;
            S[i][j] = v;
        }
    float tr = 0.0f;
    for (int i = 0; i < 16; ++i) tr += S[i][i];
    float rtr = 1.0f / tr;
    for (int i = 0; i < 16; ++i)
        for (int j = 0; j < 16; ++j) S[i][j] *= rtr;
    for (int i = 0; i < 16; ++i)
        for (int j = 0; j < 16; ++j) L[i][j] = 0.0f;
    for (int j = 0; j < 16; ++j) {
        float d = S[j][j];
        for (int k = 0; k < j; ++k) d -= L[j][k] * L[j][k];
        float dj = sqrtf(d);
        L[j][j] = dj;
        float inv = 1.0f / dj;
        for (int i = j + 1; i < 16; ++i) {
            float v = S[i][j];
            for (int k = 0; k < j; ++k) v -= L[i][k] * L[j][k];
            L[i][j] = v * inv;
        }
    }
    float* W = ws + 272;
    for (int c = 0; c < 16; ++c)
        for (int i = 0; i < 16; ++i) {
            float v;
            if (i < c) v = 0.0f;
            else {
                v = (i == c) ? 1.0f : 0.0f;
                for (int k = c; k < i; ++k) v -= L[i][k] * W[k * 16 + c];
                v /= L[i][i];
            }
            W[i * 16 + c] = v;
        }
}

// ---------------------------------------------------------------------------
// Pass 3: out = wm @ X.  Per 16-column tile: 8 coalesced b32 loads (64 B per
// half-wave per row), 4 chained WMMAs, 8 coalesced b32 stores.
// ---------------------------------------------------------------------------
__global__ void BD_apply(const float* __restrict__ x, float* __restrict__ out,
                         const float* __restrict__ ws) {
    const int tid  = threadIdx.x;
    const int lane = tid & 31;
    const int wib  = __builtin_amdgcn_readfirstlane(tid >> 5);
    const int wave = blockIdx.x * 8 + wib;
    const int n    = wave / WAVES_PER_N;
    const int tb   = wave - n * WAVES_PER_N;

    const int g  = lane & 15;
    const int hk = (lane < 16) ? 0 : 2;      // K half for A and B layouts
    const float* wm = ws + 272;

    // A blocks: wm[:, 4kc .. 4kc+3], resident for the whole kernel
    v2f a0, a1, a2, a3;
    a0.x = wm[g * 16 + 0  + hk]; a0.y = wm[g * 16 + 1  + hk];
    a1.x = wm[g * 16 + 4  + hk]; a1.y = wm[g * 16 + 5  + hk];
    a2.x = wm[g * 16 + 8  + hk]; a2.y = wm[g * 16 + 9  + hk];
    a3.x = wm[g * 16 + 12 + hk]; a3.y = wm[g * 16 + 13 + hk];

    const size_t nbase = (size_t)n * GRP * CHW;
    const float* xb = x + nbase;
    float* ob = out + nbase;
    const int col0 = tb * (TILES_PER_W * 16) + g;
    const int r0 = hk;                        // B vgpr0 row offset: 0 or 2

#pragma unroll 2
    for (int it = 0; it < TILES_PER_W; ++it) {
        const int col = col0 + it * 16;
        // B blocks: VGPR0 = row (4kc+r0), VGPR1 = row (4kc+r0+1), column col
        v2f b0, b1, b2, b3;
        b0.x = xb[(size_t)(0  + r0) * CHW + col]; b0.y = xb[(size_t)(1  + r0) * CHW + col];
        b1.x = xb[(size_t)(4  + r0) * CHW + col]; b1.y = xb[(size_t)(5  + r0) * CHW + col];
        b2.x = xb[(size_t)(8  + r0) * CHW + col]; b2.y = xb[(size_t)(9  + r0) * CHW + col];
        b3.x = xb[(size_t)(12 + r0) * CHW + col]; b3.y = xb[(size_t)(13 + r0) * CHW + col];

        v8f c = {};
        c = __builtin_amdgcn_wmma_f32_16x16x4_f32(false, a0, false, b0, (short)0, c, false, false);
        c = __builtin_amdgcn_wmma_f32_16x16x4_f32(false, a1, false, b1, (short)0, c, false, false);
        c = __builtin_amdgcn_wmma_f32_16x16x4_f32(false, a2, false, b2, (short)0, c, false, false);
        c = __builtin_amdgcn_wmma_f32_16x16x4_f32(false, a3, false, b3, (short)0, c, false, false);

#pragma unroll
        for (int r = 0; r < 8; ++r) {
            int row = r + ((lane < 16) ? 0 : 8);
            ob[(size_t)row * CHW + col] = c[r];
        }
    }
}

extern "C" void kernel_launch(void* const* d_in, const int* in_sizes, int n_in,
                              void* d_out, int out_size, void* d_ws, size_t ws_size,
                              hipStream_t stream) {
    const float* x  = (const float*)d_in[0];
    float* out      = (float*)d_out;
    float* ws       = (float*)d_ws;

    BD_ws_zero<<<1, 512, 0, stream>>>(ws);
    BD_gram  <<<BLOCKS, 256, 0, stream>>>(x, ws);   // 14336 waves, 224 cols each
    BD_chol  <<<1, 32, 0, stream>>>(ws);
    BD_apply <<<BLOCKS, 256, 0, stream>>>(x, out, ws);
}